// ChamferLoss_16509854286171
// MI455X (gfx1250) — compile-verified
//
#include <hip/hip_runtime.h>
#include <math.h>

// CDNA5 / gfx1250 Chamfer loss.
// Pairwise dot products via V_WMMA_F32_16X16X4_F32 (K=3 padded to 4),
// row-mins (template->source) kept as u32 float-bits in registers (d2 clamped
// >= 0 so unsigned order == float order -> v_min_u32, no NaN canonicalization),
// col-mins (source->template) combined in LDS via ds_min_u32.
// B-tile loads are software-pipelined one tile ahead, prefetch two ahead.

typedef __attribute__((ext_vector_type(2))) float v2f;
typedef __attribute__((ext_vector_type(8))) float v8f;

#define NPTS 8192
#define NBATCH 4
#define WAVES_PER_BLOCK 8
#define COLS_PER_BLOCK 2048
#define COL_TILES (COLS_PER_BLOCK / 16)        // 128
#define ROWS_PER_BLOCK (WAVES_PER_BLOCK * 16)  // 128
#define INF_BITS 0x7F800000u

__device__ __forceinline__ unsigned int umin32(unsigned int a, unsigned int b) {
    return a < b ? a : b;
}

__global__ void chamfer_init_kernel(unsigned int* __restrict__ p, int n) {
    int i = blockIdx.x * blockDim.x + threadIdx.x;
    if (i < n) p[i] = INF_BITS;
}

__global__ __launch_bounds__(256) void chamfer_main_kernel(
    const float* __restrict__ tpl,      // (B, N, 3)
    const float* __restrict__ src,      // (B, M, 3)
    unsigned int* __restrict__ rowmin,  // (B*N) min_m d2, f32 bits
    unsigned int* __restrict__ colmin)  // (B*M) min_n d2, f32 bits
{
    __shared__ unsigned int s_colmin[COLS_PER_BLOCK];

    const int tid = threadIdx.x;
    for (int i = tid; i < COLS_PER_BLOCK; i += 256) s_colmin[i] = INF_BITS;
    __syncthreads();

    const int lane = tid & 31;
    const int wave = tid >> 5;
    const int half = lane >> 4;   // 0: lanes 0-15 (K0/K1, C rows 0-7), 1: lanes 16-31 (K2/K3, rows 8-15)
    const int l16  = lane & 15;

    const int b       = blockIdx.z;
    const int row0    = blockIdx.x * ROWS_PER_BLOCK + wave * 16;  // wave's 16-row A tile
    const int colbase = blockIdx.y * COLS_PER_BLOCK;

    const float* tb = tpl + (size_t)b * NPTS * 3;
    const float* sb = src + (size_t)b * NPTS * 3;

    // ---- Load A tile once per wave: row = row0 + (lane&15) ----
    const int ar = row0 + l16;
    const float ax = tb[ar * 3 + 0];
    const float ay = tb[ar * 3 + 1];
    const float az = tb[ar * 3 + 2];
    const float a2 = ax * ax + ay * ay + az * az;

    v2f A;
    A.x = half ? az : ax;   // VGPR0: K0 (x) / K2 (z)
    A.y = half ? 0.f : ay;  // VGPR1: K1 (y) / K3 (zero pad)

    // Row norms for this lane's 8 C-matrix rows (rows i / i+8 by half).
    float a2r[8];
#pragma unroll
    for (int i = 0; i < 8; ++i)
        a2r[i] = __shfl(a2, half ? (i + 8) : i, 32);

    unsigned int rminb[8];
#pragma unroll
    for (int i = 0; i < 8; ++i) rminb[i] = INF_BITS;

    // ---- Software-pipelined sweep over 128 column tiles ----
    const float* sp = sb + (size_t)(colbase + l16) * 3;  // this lane's column, tile 0
    float bx = sp[0], by = sp[1], bz = sp[2];

    for (int ct = 0; ct < COL_TILES; ++ct) {
        // Preload next tile (branchless clamp on the final iteration: harmless re-read).
        const float* spn = (ct + 1 < COL_TILES) ? (sp + 48) : sp;
        __builtin_prefetch(spn + 48, 0, 3);   // global_prefetch_b8 two tiles ahead
        const float nbx = spn[0];
        const float nby = spn[1];
        const float nbz = spn[2];

        const float b2 = bx * bx + by * by + bz * bz;
        v2f Bm;
        Bm.x = half ? bz : bx;
        Bm.y = half ? 0.f : by;

        v8f C = {};
        // ab[row][col] tile: A(16x4,f32) x B(4x16,f32)
        C = __builtin_amdgcn_wmma_f32_16x16x4_f32(
                /*neg_a=*/false, A, /*neg_b=*/false, Bm,
                /*c_mod=*/(short)0, C, /*reuse_a=*/false, /*reuse_b=*/false);

        unsigned int cminb = INF_BITS;
#pragma unroll
        for (int i = 0; i < 8; ++i) {
            float d2 = fmaf(-2.0f, C[i], a2r[i] + b2);
            d2 = fmaxf(d2, 0.0f);                    // nonneg => u32 order == float order
            const unsigned int db = __float_as_uint(d2);
            rminb[i] = umin32(rminb[i], db);         // v_min_u32, no canonicalization
            cminb    = umin32(cminb, db);
        }
        atomicMin(&s_colmin[ct * 16 + l16], cminb);  // ds_min_u32

        bx = nbx; by = nby; bz = nbz;
        sp = spn;
    }

    // ---- Row mins: reduce across the 16-lane group, flush to global ----
#pragma unroll
    for (int i = 0; i < 8; ++i) {
        unsigned int v = rminb[i];
        v = umin32(v, (unsigned int)__shfl_xor((int)v, 1, 32));
        v = umin32(v, (unsigned int)__shfl_xor((int)v, 2, 32));
        v = umin32(v, (unsigned int)__shfl_xor((int)v, 4, 32));
        v = umin32(v, (unsigned int)__shfl_xor((int)v, 8, 32));
        if (l16 == 0) {
            const int r = row0 + i + half * 8;
            atomicMin(&rowmin[b * NPTS + r], v);
        }
    }

    // ---- Col mins: block-combined in LDS, one global atomic per column ----
    __syncthreads();
    for (int i = tid; i < COLS_PER_BLOCK; i += 256)
        atomicMin(&colmin[b * NPTS + colbase + i], s_colmin[i]);
}

__global__ __launch_bounds__(256) void chamfer_reduce_kernel(
    const unsigned int* __restrict__ mins, int n, float* __restrict__ out)
{
    __shared__ float s[256];
    float acc = 0.f;
    for (int i = threadIdx.x; i < n; i += 256) {
        const float d2 = __uint_as_float(mins[i]);
        acc += sqrtf(fmaxf(d2, 1e-12f));
    }
    s[threadIdx.x] = acc;
    __syncthreads();
    for (int off = 128; off > 0; off >>= 1) {
        if (threadIdx.x < off) s[threadIdx.x] += s[threadIdx.x + off];
        __syncthreads();
    }
    if (threadIdx.x == 0)
        out[0] = s[0] / (float)(NBATCH * NPTS);  // (sum_rows + sum_cols) / (B*N)
}

extern "C" void kernel_launch(void* const* d_in, const int* in_sizes, int n_in,
                              void* d_out, int out_size, void* d_ws, size_t ws_size,
                              hipStream_t stream) {
    (void)in_sizes; (void)n_in; (void)out_size; (void)ws_size;
    const float* tpl = (const float*)d_in[0];
    const float* src = (const float*)d_in[1];
    float* out = (float*)d_out;

    unsigned int* mins = (unsigned int*)d_ws;  // rowmin: [0, B*N), colmin: [B*N, 2*B*N)
    const int total = 2 * NBATCH * NPTS;       // 65536 u32 = 256 KB of d_ws

    chamfer_init_kernel<<<(total + 255) / 256, 256, 0, stream>>>(mins, total);

    dim3 grid(NPTS / ROWS_PER_BLOCK,   // 64 row groups
              NPTS / COLS_PER_BLOCK,   // 4 column chunks
              NBATCH);                 // 4 batches -> 1024 blocks x 8 waves
    chamfer_main_kernel<<<grid, 256, 0, stream>>>(tpl, src, mins, mins + NBATCH * NPTS);

    chamfer_reduce_kernel<<<1, 256, 0, stream>>>(mins, total, out);
}